// SelectedMSELoss_33208687133246
// MI455X (gfx1250) — compile-verified
//
#include <hip/hip_runtime.h>

// SelectedMSELoss for (B=256, C=224*224) f32 tensors on gfx1250 (MI455X).
//
// Memory-bound streaming reduction (~103 MB read -> ~4.4 us HBM floor at
// 23.3 TB/s; fits in 192 MB L2 for graph replays). The batch-axis column
// reduction is performed with V_WMMA_F32_16X16X4_F32 using an all-ones
// A-matrix, so D[m][n] = sum_k B[k][n] independent of the exact K-row->slot
// mapping of the B operand (layout-robust). Column n's sum lands in lane
// (n mod 16) of every D row; we read D VGPR0.

typedef __attribute__((ext_vector_type(2))) float v2f;
typedef __attribute__((ext_vector_type(8))) float v8f;

#define NUM_CLASSES (224 * 224)   // 50176
#define BATCH       256
#define COLS_PER_WAVE  16
#define WAVES_PER_BLOCK 8
#define COLS_PER_BLOCK (COLS_PER_WAVE * WAVES_PER_BLOCK)   // 128
#define NUM_BLOCKS (NUM_CLASSES / COLS_PER_BLOCK)          // 392 exact

__global__ void selmse_zero_kernel(float* out) {
    out[0] = 0.0f;
}

__global__ __launch_bounds__(256)
void selmse_main_kernel(const float* __restrict__ in,
                        const float* __restrict__ tg,
                        float* __restrict__ out) {
    const int tid  = threadIdx.x;
    const int lane = tid & 31;
    const int wave = tid >> 5;
    const int half = lane >> 4;                 // 0: lanes 0-15, 1: lanes 16-31
    const int col  = blockIdx.x * COLS_PER_BLOCK + wave * COLS_PER_WAVE + (lane & 15);

    const float* __restrict__ inc = in + col;   // walk this column down the batch
    const float* __restrict__ tgc = tg + col;

    v8f csum = {};                              // masked-loss column sums
    v8f ccnt = {};                              // mask counts
    v2f ones; ones[0] = 1.0f; ones[1] = 1.0f;   // A = 16x4 all-ones (both A VGPRs)

    // 4 batch rows per WMMA: B operand vgpr v, lane L holds row r0 + v + 2*(L>=16)
    // (any bijection of the 4 rows onto the 4 slots yields the same column sums
    //  because A is all-ones). Lanes 0-15 and 16-31 each read one contiguous
    //  64-byte segment per load -> fully consumed cachelines.
#pragma unroll 4
    for (int r0 = 0; r0 < BATCH; r0 += 4) {
        const int ra = r0 + 2 * half;
        const int rb = ra + 1;
        float i0 = inc[(size_t)ra * NUM_CLASSES];
        float t0 = tgc[(size_t)ra * NUM_CLASSES];
        float i1 = inc[(size_t)rb * NUM_CLASSES];
        float t1 = tgc[(size_t)rb * NUM_CLASSES];

        float m0 = (t0 > 0.0f) ? 1.0f : 0.0f;
        float m1 = (t1 > 0.0f) ? 1.0f : 0.0f;
        float d0 = i0 - t0;
        float d1 = i1 - t1;

        v2f be; be[0] = d0 * d0 * m0; be[1] = d1 * d1 * m1;
        v2f bm; bm[0] = m0;           bm[1] = m1;

        // 8-arg form: (neg_a, A, neg_b, B, c_mod, C, reuse_a, reuse_b)
        csum = __builtin_amdgcn_wmma_f32_16x16x4_f32(false, ones, false, be,
                                                     (short)0, csum, false, false);
        ccnt = __builtin_amdgcn_wmma_f32_16x16x4_f32(false, ones, false, bm,
                                                     (short)0, ccnt, false, false);
    }

    // D VGPR0: every lane L holds the column total for column base + (L & 15)
    // (lanes 0-15 are row M=0, lanes 16-31 are row M=8 -- all D rows equal).
    float s = csum[0];
    float n = ccnt[0];
    float mean = (n > 0.0f) ? (s / n) : 0.0f;
    // Each column is present in two lanes (L and L+16); let only half 0 emit.
    float contrib = (half == 0) ? mean * (1.0f / (float)NUM_CLASSES) : 0.0f;

    __shared__ float bsum;
    if (tid == 0) bsum = 0.0f;
    __syncthreads();
    atomicAdd(&bsum, contrib);                  // ds_add_f32
    __syncthreads();
    if (tid == 0) atomicAdd(out, bsum);         // global_atomic_add_f32
}

extern "C" void kernel_launch(void* const* d_in, const int* in_sizes, int n_in,
                              void* d_out, int out_size, void* d_ws, size_t ws_size,
                              hipStream_t stream) {
    const float* in = (const float*)d_in[0];
    const float* tg = (const float*)d_in[1];
    float* out = (float*)d_out;

    // Zero the scalar accumulator every launch (harness poisons once, never
    // re-zeroes between graph replays).
    hipLaunchKernelGGL(selmse_zero_kernel, dim3(1), dim3(1), 0, stream, out);

    hipLaunchKernelGGL(selmse_main_kernel, dim3(NUM_BLOCKS), dim3(256), 0, stream,
                       in, tg, out);
}